// RetinaNetLoss_46763603919170
// MI455X (gfx1250) — compile-verified
//
#include <hip/hip_runtime.h>
#include <hip/hip_bf16.h>

#define BATCH 8
#define NA    120000
#define NM    64

typedef unsigned int u32;
typedef __attribute__((ext_vector_type(4))) u32 u32x4;
typedef __attribute__((ext_vector_type(8))) u32 u32x8;

// ---------------------------------------------------------------------------
// Kernel 1: zero the 24-float workspace accumulators
// ws[0..7]=cls_sum[b], ws[8..15]=reg_sum[b], ws[16..23]=pos_cnt[b]
// ---------------------------------------------------------------------------
__global__ void retina_init(float* __restrict__ ws) {
    int t = threadIdx.x;
    if (t < 24) ws[t] = 0.0f;
}

// ---------------------------------------------------------------------------
// Kernel 2: per-anchor IoU max/argmax vs <=64 GT boxes + focal + smooth-L1.
// GT table staged with BOTH CDNA5 DMA paths (identical bytes, benign overlap):
//   - TDM tensor_load_to_lds (TENSORcnt), one instruction from wave 0
//   - per-lane global_load_async_to_lds_b128 (ASYNCcnt)
// ---------------------------------------------------------------------------
__global__ __launch_bounds__(256) void retina_main(
    const float* __restrict__ gt,     // (B, 64, 5) cx cy w h label
    const float* __restrict__ pcls,   // (B, A, 1)
    const float* __restrict__ preg,   // (B, A, 4)
    const float* __restrict__ anc,    // (1, A, 4) x1 y1 x2 y2
    float* __restrict__ ws)
{
    __shared__ float  s_raw[NM * 5];   // 1280 B DMA target
    __shared__ float4 s_box[NM];       // x1 y1 x2 y2
    __shared__ float  s_ar[NM];        // corner-derived area
    __shared__ int    s_ngt;

    const int t = threadIdx.x;
    const int b = blockIdx.y;
    const int a = blockIdx.x * 256 + t;

    if (t == 0) s_ngt = 0;

    const unsigned long long ga =
        (unsigned long long)(const void*)(gt + (long)b * NM * 5);
    const unsigned lds_base = (unsigned)(unsigned long long)(const void*)&s_raw[0];

    // ---- TDM: one 320-element (1280 B) 1-row tile, Global -> LDS ----
    if (t < 32) {                                   // wave 0 only (uniform)
        u32x4 g0 = { 1u,                            // count=1, user desc
                     lds_base,                      // lds_addr
                     (u32)ga,                       // global_addr[31:0]
                     (u32)((ga >> 32) & 0x1FFFFFFu) | (2u << 30) }; // [56:32]|type=2
        u32x8 g1 = { 2u << 16,                      // data_size=4B
                     320u << 16,                    // tensor_dim0=320 (lo16 in s1 hi)
                     1u << 16,                      // tensor_dim0 hi=0, tensor_dim1=1
                     320u << 16,                    // tensor_dim1 hi=0, tile_dim0=320
                     1u,                            // tile_dim1=1, tile_dim2=0
                     320u,                          // tensor_dim0_stride=320 (lo32)
                     0u, 0u };                      // stride hi / dim1_stride
        asm volatile("tensor_load_to_lds %0, %1" :: "s"(g0), "s"(g1) : "memory");
        __builtin_amdgcn_s_wait_tensorcnt(0);
    }
    // ---- async per-lane B128 copies of the same bytes (ASYNCcnt path) ----
    if (t < 80) {
        unsigned ldsa = lds_base + (unsigned)t * 16u;
        unsigned voff = (unsigned)t * 16u;
        asm volatile("global_load_async_to_lds_b128 %0, %1, %2"
                     :: "v"(ldsa), "v"(voff), "s"(ga) : "memory");
    }
    asm volatile("s_wait_asynccnt 0x0" ::: "memory");
    __syncthreads();

    // ---- derive corners/areas; count valid prefix (valid rows are a prefix) ----
    if (t < NM) {
        float cx = s_raw[t * 5 + 0], cy = s_raw[t * 5 + 1];
        float w  = s_raw[t * 5 + 2], h  = s_raw[t * 5 + 3];
        float lab = s_raw[t * 5 + 4];
        float x1 = cx - 0.5f * w, y1 = cy - 0.5f * h;
        float x2 = cx + 0.5f * w, y2 = cy + 0.5f * h;
        s_box[t] = make_float4(x1, y1, x2, y2);
        s_ar[t]  = (x2 - x1) * (y2 - y1);
        if (lab >= 0.0f) atomicMax(&s_ngt, t + 1);
    }
    __syncthreads();
    const int ngt = s_ngt;                          // wave-uniform trip count

    float clsv = 0.0f, regv = 0.0f, posv = 0.0f;

    if (a < NA) {
        // issue global loads before the loop so latency hides under VALU work
        const float4 av = *(const float4*)(anc + (long)a * 4);
        const float  p  = pcls[(long)b * NA + a];
        const float4 rr = *(const float4*)(preg + ((long)b * NA + a) * 4);

        const float ax1 = av.x, ay1 = av.y, ax2 = av.z, ay2 = av.w;
        const float aw = ax2 - ax1, ah = ay2 - ay1;
        const float acx = ax1 + 0.5f * aw, acy = ay1 + 0.5f * ah;
        const float area_a = aw * ah;

        // running max of inter/den tracked as a ratio pair (den > 0 always)
        float bn = -1.0f, bd = 1.0f;
        int   arg = 0;
        for (int m = 0; m < ngt; ++m) {
            float4 g = s_box[m];                    // ds_load_b128 broadcast
            float lx = fmaxf(ax1, g.x);
            float ly = fmaxf(ay1, g.y);
            float rx = fminf(ax2, g.z);
            float ry = fminf(ay2, g.w);
            float wi = fmaxf(rx - lx, 0.0f);
            float hi = fmaxf(ry - ly, 0.0f);
            float inter = wi * hi;
            float den = area_a + s_ar[m] - inter;
            if (inter * bd > bn * den) { bn = inter; bd = den; arg = m; }
        }

        const bool pos = (2.0f * bn >= bd);         // iou_max >= 0.5
        const bool neg = (5.0f * bn <  2.0f * bd);  // iou_max <  0.4
        posv = pos ? 1.0f : 0.0f;

        // focal loss, C==1: valid labels are all 0 => one-hot == 1 for pos
        if (pos) {
            float fw = 1.0f - p;
            clsv = 0.25f * fw * fw * (-__logf(p));
        } else if (neg) {
            clsv = 0.75f * p * p * (-__logf(1.0f - p));
        }

        // smooth-L1 regression (positives only)
        if (pos) {
            float gcx = s_raw[arg * 5 + 0], gcy = s_raw[arg * 5 + 1];
            float gw  = fmaxf(s_raw[arg * 5 + 2], 1.0f);
            float gh  = fmaxf(s_raw[arg * 5 + 3], 1.0f);
            float t0 = ((gcx - acx) / aw) * 10.0f;  // / STD 0.1
            float t1 = ((gcy - acy) / ah) * 10.0f;
            float t2 = __logf(gw / aw) * 5.0f;      // / STD 0.2
            float t3 = __logf(gh / ah) * 5.0f;
            float d0 = fabsf(t0 - rr.x), d1 = fabsf(t1 - rr.y);
            float d2 = fabsf(t2 - rr.z), d3 = fabsf(t3 - rr.w);
            const float beta = 1.0f / 9.0f, hb = 0.5f / 9.0f;
            float rl = 0.0f;
            rl += (d0 <= beta) ? 4.5f * d0 * d0 : d0 - hb;
            rl += (d1 <= beta) ? 4.5f * d1 * d1 : d1 - hb;
            rl += (d2 <= beta) ? 4.5f * d2 * d2 : d2 - hb;
            rl += (d3 <= beta) ? 4.5f * d3 * d3 : d3 - hb;
            regv = rl;
        }
    }

    // ---- wave32 tree reduction, one atomic per wave per accumulator ----
    for (int off = 16; off > 0; off >>= 1) {
        clsv += __shfl_down(clsv, off, 32);
        regv += __shfl_down(regv, off, 32);
        posv += __shfl_down(posv, off, 32);
    }
    if ((t & 31) == 0) {
        atomicAdd(&ws[b],       clsv);
        atomicAdd(&ws[8  + b],  regv);
        atomicAdd(&ws[16 + b],  posv);
    }
}

// ---------------------------------------------------------------------------
// Kernel 3: per-image normalization + batch mean -> d_out[0..1]
// ---------------------------------------------------------------------------
__global__ void retina_final(const float* __restrict__ ws, float* __restrict__ out) {
    if (threadIdx.x == 0) {
        float cs = 0.0f, rs = 0.0f;
        for (int b = 0; b < BATCH; ++b) {
            float np = ws[16 + b];
            cs += ws[b] / fmaxf(np, 1.0f);
            rs += (np > 0.0f) ? ws[8 + b] / fmaxf(np * 4.0f, 1.0f) : 0.0f;
        }
        out[0] = cs * (1.0f / BATCH);
        out[1] = rs * (1.0f / BATCH);
    }
}

extern "C" void kernel_launch(void* const* d_in, const int* in_sizes, int n_in,
                              void* d_out, int out_size, void* d_ws, size_t ws_size,
                              hipStream_t stream) {
    const float* gt   = (const float*)d_in[0];   // (8, 64, 5)
    const float* pcls = (const float*)d_in[1];   // (8, 120000, 1)
    const float* preg = (const float*)d_in[2];   // (8, 120000, 4)
    const float* anc  = (const float*)d_in[3];   // (1, 120000, 4)
    float* ws  = (float*)d_ws;
    float* out = (float*)d_out;

    retina_init<<<1, 32, 0, stream>>>(ws);
    dim3 grid((NA + 255) / 256, BATCH);
    retina_main<<<grid, 256, 0, stream>>>(gt, pcls, preg, anc, ws);
    retina_final<<<1, 32, 0, stream>>>(ws, out);
}